// GNNModel_24850680775340
// MI455X (gfx1250) — compile-verified
//
#include <hip/hip_runtime.h>

typedef __attribute__((ext_vector_type(2))) float v2f;
typedef __attribute__((ext_vector_type(8))) float v8f;

// ---------------------------------------------------------------- utilities

__global__ void gcn_fill_f32(float* __restrict__ p, float v, long long n) {
    long long i = (long long)blockIdx.x * blockDim.x + threadIdx.x;
    if (i < n) p[i] = v;
}

__global__ void gcn_deg_count(const long long* __restrict__ dst,
                              float* __restrict__ deg, long long E) {
    long long i = (long long)blockIdx.x * blockDim.x + threadIdx.x;
    if (i < E) atomicAdd(&deg[(int)dst[i]], 1.0f);
}

__global__ void gcn_rsqrt(float* __restrict__ d, int n) {
    int i = blockIdx.x * blockDim.x + threadIdx.x;
    if (i < n) {
        float v = d[i];
        d[i] = (v > 0.0f) ? rsqrtf(v) : 0.0f;
    }
}

// ------------------------------------------------- WMMA f32 GEMM (row-major)
// C[M,N] = A[M,K] * W[K,N] (+ bias). One 16x16 output tile per wave32.
// K must be a multiple of 4. Early exit is wave-uniform so EXEC stays all-1s
// for every executed V_WMMA (ISA requirement).
__global__ void gcn_gemm_wmma_f32(const float* __restrict__ A,
                                  const float* __restrict__ W,
                                  const float* __restrict__ bias,
                                  float* __restrict__ C,
                                  int M, int N, int K) {
    int wave  = (int)((blockIdx.x * blockDim.x + threadIdx.x) >> 5);
    int lane  = (int)(threadIdx.x & 31);
    int tilesN = N >> 4;
    int tiles  = (M >> 4) * tilesN;
    if (wave >= tiles) return;              // uniform across the wave

    int tm   = wave / tilesN;
    int tn   = wave % tilesN;
    int row0 = tm << 4;
    int col0 = tn << 4;
    int half = lane >> 4;                   // 0: lanes 0-15, 1: lanes 16-31
    int l16  = lane & 15;

    const float* arow = A + (size_t)(row0 + l16) * K;

    v8f acc = {};
    for (int k = 0; k < K; k += 4) {
        // A 16x4 fragment: VGPR0 = K=(k+2*half), VGPR1 = K=(k+2*half+1)
        int ka = k + (half << 1);
        v2f a, b;
        a.x = arow[ka];
        a.y = arow[ka + 1];
        // B 4x16 fragment mirrors the A K-distribution, N striped across lanes
        b.x = W[(size_t)ka * N + col0 + l16];
        b.y = W[(size_t)(ka + 1) * N + col0 + l16];
        acc = __builtin_amdgcn_wmma_f32_16x16x4_f32(
            /*neg_a=*/false, a, /*neg_b=*/false, b,
            /*c_mod=*/(short)0, acc, /*reuse_a=*/false, /*reuse_b=*/false);
    }

    // C/D layout: VGPR v -> M = row0 + v + 8*half, N = col0 + l16
#pragma unroll
    for (int v = 0; v < 8; ++v) {
        int m = row0 + v + (half << 3);
        int n = col0 + l16;
        float val = acc[v];
        if (bias) val += bias[n];
        C[(size_t)m * N + n] = val;
    }
}

// ------------------------------------------------- edge gather/scale/scatter
// One (edge, 4-feature chunk) per thread: 32 threads per edge for F=128.
__global__ void gcn_scatter(const long long* __restrict__ src,
                            const long long* __restrict__ dst,
                            const float* __restrict__ dinv,
                            const float* __restrict__ h,
                            float* __restrict__ agg,
                            long long E, int F) {
    long long idx    = (long long)blockIdx.x * blockDim.x + threadIdx.x;
    int       chunks = F >> 2;              // float4 chunks per edge
    long long total  = E * chunks;
    if (idx >= total) return;
    long long e  = idx / chunks;
    int       f4 = (int)(idx % chunks) << 2;

    int s = (int)src[e];
    int d = (int)dst[e];
    float nrm = dinv[s] * dinv[d];

    const float4 hv = *(const float4*)(h + (size_t)s * F + f4);
    float* out = agg + (size_t)d * F + f4;
    atomicAdd(out + 0, hv.x * nrm);
    atomicAdd(out + 1, hv.y * nrm);
    atomicAdd(out + 2, hv.z * nrm);
    atomicAdd(out + 3, hv.w * nrm);
}

// ------------------------------------- self-loop + bias + ReLU epilogue
__global__ void gcn_finish(const float* __restrict__ agg,
                           const float* __restrict__ h,
                           const float* __restrict__ dinv,
                           const float* __restrict__ bias,
                           float* __restrict__ out,
                           int n, int F) {
    long long idx = (long long)blockIdx.x * blockDim.x + threadIdx.x;
    long long tot = (long long)n * F;
    if (idx >= tot) return;
    int i = (int)(idx / F);
    int f = (int)(idx % F);
    float di = dinv[i];
    float v  = agg[idx] + h[idx] * di * di + bias[f];
    out[idx] = fmaxf(v, 0.0f);
}

// ---------------------------------------------------------------- launcher

extern "C" void kernel_launch(void* const* d_in, const int* in_sizes, int n_in,
                              void* d_out, int out_size, void* d_ws, size_t ws_size,
                              hipStream_t stream) {
    const float*     x   = (const float*)d_in[0];
    const long long* ei  = (const long long*)d_in[1];   // int64 edge_index [2,E]
    const float*     w1  = (const float*)d_in[2];
    const float*     b1  = (const float*)d_in[3];
    const float*     w2  = (const float*)d_in[4];
    const float*     b2  = (const float*)d_in[5];
    const float*     w3  = (const float*)d_in[6];
    const float*     b3  = (const float*)d_in[7];
    const float*     wo  = (const float*)d_in[8];
    const float*     bo  = (const float*)d_in[9];

    const int  F = 128;                     // hidden width
    const int  n = in_sizes[0] / F;         // 10000 nodes
    const long long E = in_sizes[1] / 2;    // 640000 edges

    // workspace layout (all L2-resident: ~21 MB total)
    size_t featBytes = ((size_t)n * F * sizeof(float) + 255) & ~(size_t)255;
    char*  ws   = (char*)d_ws;
    float* dinv = (float*)ws;                               // n floats
    float* H    = (float*)(ws + 65536);                     // gemm output
    float* AGG  = (float*)(ws + 65536 + 1 * featBytes);     // scatter target
    float* F0   = (float*)(ws + 65536 + 2 * featBytes);     // feat ping
    float* F1   = (float*)(ws + 65536 + 3 * featBytes);     // feat pong

    const int TB = 256;
    long long nf = (long long)n * F;

    // ---- degrees -> dinv (self-loop contributes the initial 1.0)
    gcn_fill_f32<<<(unsigned)((n + TB - 1) / TB), TB, 0, stream>>>(dinv, 1.0f, n);
    gcn_deg_count<<<(unsigned)((E + TB - 1) / TB), TB, 0, stream>>>(ei + E, dinv, E);
    gcn_rsqrt<<<(unsigned)((n + TB - 1) / TB), TB, 0, stream>>>(dinv, n);

    const float* Ws[3] = {w1, w2, w3};
    const float* Bs[3] = {b1, b2, b3};
    const float* feat  = x;

    // gemm launch geometry: one wave32 per 16x16 tile
    auto gemm_blocks = [&](int M, int N) -> unsigned {
        long long waves = (long long)(M / 16) * (N / 16);
        return (unsigned)((waves * 32 + TB - 1) / TB);
    };

    long long scatter_threads = E * (F / 4);
    unsigned  scatter_blocks  = (unsigned)((scatter_threads + TB - 1) / TB);

    for (int l = 0; l < 3; ++l) {
        gcn_gemm_wmma_f32<<<gemm_blocks(n, F), TB, 0, stream>>>(
            feat, Ws[l], nullptr, H, n, F, F);
        gcn_fill_f32<<<(unsigned)((nf + TB - 1) / TB), TB, 0, stream>>>(AGG, 0.0f, nf);
        gcn_scatter<<<scatter_blocks, TB, 0, stream>>>(ei, ei + E, dinv, H, AGG, E, F);
        float* fout = (l & 1) ? F1 : F0;
        gcn_finish<<<(unsigned)((nf + TB - 1) / TB), TB, 0, stream>>>(
            AGG, H, dinv, Bs[l], fout, n, F);
        feat = fout;
    }

    // final linear: [n,128] x [128,64] + bias -> d_out
    gcn_gemm_wmma_f32<<<gemm_blocks(n, 64), TB, 0, stream>>>(
        feat, wo, bo, (float*)d_out, n, 64, F);
}